// MatrixGatedDeltaNetBlock_9088150798903
// MI455X (gfx1250) — compile-verified
//
#include <hip/hip_runtime.h>
#include <hip/hip_bf16.h>
#include <math.h>

typedef __attribute__((ext_vector_type(2))) float v2f;
typedef __attribute__((ext_vector_type(8))) float v8f;

#define B_   2
#define S_   1024
#define HID_ 1024
#define NH_  16
#define D_   64
#define CLIP_ 5.0f
#define EPS_  1e-6f

// ---------------------------------------------------------------------------
// fp32 GEMM: C[M,N] = A[M,K] @ B[K,N], row-major, M%128==0, N%64==0, K%32==0.
// Block = 256 threads (8 waves) computes a 128x64 C tile; each wave computes
// a 32x32 sub-tile as 2x2 V_WMMA_F32_16X16X4_F32 accumulators.
// A staged row-major As[m][k]; B staged TRANSPOSED Bst[n][k] so both A and B
// WMMA fragments are contiguous 8B-aligned v2f LDS loads (ds_load_b64), no
// per-fragment register shuffling. Next K-chunk is prefetched
// (global_prefetch_b8) while the current chunk's 32 WMMAs issue.
// ---------------------------------------------------------------------------
__global__ __launch_bounds__(256)
void gemm_f32_wmma(const float* __restrict__ A, const float* __restrict__ Bm,
                   float* __restrict__ C, int M, int N, int K) {
    __shared__ float As[128][36];   // 128x32 A tile, row padded to 36 (16B align)
    __shared__ float Bst[64][34];   // 32x64 B tile stored transposed, pad 34 (8B align)

    const int tid  = threadIdx.x;
    const int gm0  = blockIdx.y * 128;
    const int gn0  = blockIdx.x * 64;
    const int wave = tid >> 5;
    const int lane = tid & 31;
    const int lo   = lane & 15;
    const int hi   = lane >> 4;
    const int wm   = wave >> 1;   // 0..3 -> 32-row band
    const int wn   = wave & 1;    // 0..1 -> 32-col band

    // cooperative loader indices
    const int ar = tid >> 1;           // 0..127 (A row in tile)
    const int ac = (tid & 1) * 16;     // 0 or 16
    const int br = tid >> 3;           // 0..31  (B row in tile)
    const int bc = (tid & 7) * 8;      // 0..56

    v8f acc00 = {}, acc01 = {}, acc10 = {}, acc11 = {};

    const float* aSrcBase = A  + (size_t)(gm0 + ar) * K + ac;
    const float* bSrcBase = Bm + (size_t)br * N + gn0 + bc;

    for (int k0 = 0; k0 < K; k0 += 32) {
        __syncthreads();
        {   // A tile: 128x32, 16 floats per thread
            const float* src = aSrcBase + k0;
            float4 p0 = *(const float4*)(src);
            float4 p1 = *(const float4*)(src + 4);
            float4 p2 = *(const float4*)(src + 8);
            float4 p3 = *(const float4*)(src + 12);
            *(float4*)&As[ar][ac]      = p0;
            *(float4*)&As[ar][ac + 4]  = p1;
            *(float4*)&As[ar][ac + 8]  = p2;
            *(float4*)&As[ar][ac + 12] = p3;
        }
        {   // B tile: 32x64, stored transposed (Bst[n][k] = B[k0+k][gn0+n])
            const float* src = bSrcBase + (size_t)k0 * N;
            float4 p0 = *(const float4*)(src);
            float4 p1 = *(const float4*)(src + 4);
            float t8[8] = {p0.x, p0.y, p0.z, p0.w, p1.x, p1.y, p1.z, p1.w};
            #pragma unroll
            for (int i = 0; i < 8; ++i) Bst[bc + i][br] = t8[i];
        }
        if (k0 + 32 < K) {   // prefetch next chunk (uniform branch, EXEC stays full)
            __builtin_prefetch(aSrcBase + k0 + 32, 0, 3);
            __builtin_prefetch(bSrcBase + (size_t)(k0 + 32) * N, 0, 3);
        }
        __syncthreads();

        #pragma unroll
        for (int kk = 0; kk < 8; ++kk) {
            const int krow = kk * 4 + 2 * hi;
            // A fragments: A[m=lo(+16)][krow + j]
            v2f a0 = *(const v2f*)&As[wm * 32 + lo][krow];
            v2f a1 = *(const v2f*)&As[wm * 32 + 16 + lo][krow];
            // B fragments: B[krow + j][n=lo(+16)] via transposed LDS
            v2f b0 = *(const v2f*)&Bst[wn * 32 + lo][krow];
            v2f b1 = *(const v2f*)&Bst[wn * 32 + 16 + lo][krow];
            acc00 = __builtin_amdgcn_wmma_f32_16x16x4_f32(
                false, a0, false, b0, (short)0, acc00, false, false);
            acc01 = __builtin_amdgcn_wmma_f32_16x16x4_f32(
                false, a0, false, b1, (short)0, acc01, false, false);
            acc10 = __builtin_amdgcn_wmma_f32_16x16x4_f32(
                false, a1, false, b0, (short)0, acc10, false, false);
            acc11 = __builtin_amdgcn_wmma_f32_16x16x4_f32(
                false, a1, false, b1, (short)0, acc11, false, false);
        }
    }

    // C/D layout: lane (lo,hi), VGPR r -> D[8*hi + r][lo]
    const int row0 = gm0 + wm * 32 + hi * 8;
    const int col0 = gn0 + wn * 32 + lo;
    #pragma unroll
    for (int r = 0; r < 8; ++r) {
        C[(size_t)(row0 + r) * N + col0]           = acc00[r];
        C[(size_t)(row0 + r) * N + col0 + 16]      = acc01[r];
        C[(size_t)(row0 + 16 + r) * N + col0]      = acc10[r];
        C[(size_t)(row0 + 16 + r) * N + col0 + 16] = acc11[r];
    }
}

// ---------------------------------------------------------------------------
// alpha/beta gates: per row m of x[2048,1024], out[j] = sigmoid(x_row @ W[:,j] + b[j])
// One block per row; x row staged in LDS; 2(which) x 16(j) x 8(chunk) partials.
// ---------------------------------------------------------------------------
__global__ __launch_bounds__(256)
void gates_kernel(const float* __restrict__ x,
                  const float* __restrict__ Wa, const float* __restrict__ biasA,
                  const float* __restrict__ Wb, const float* __restrict__ biasB,
                  float* __restrict__ alpha, float* __restrict__ beta) {
    __shared__ float xrow[HID_];
    __shared__ float part[2][16][8];
    const int m   = blockIdx.x;
    const int tid = threadIdx.x;

    const float* xp = x + (size_t)m * HID_;
    for (int i = tid; i < HID_; i += 256) xrow[i] = xp[i];
    __syncthreads();

    const int j     = tid & 15;
    const int which = (tid >> 4) & 1;
    const int chunk = tid >> 5;                 // 0..7
    const float* W  = which ? Wb : Wa;
    const int kbase = chunk * 128;
    float acc = 0.0f;
    for (int k = 0; k < 128; ++k)
        acc += xrow[kbase + k] * W[(size_t)(kbase + k) * NH_ + j];
    part[which][j][chunk] = acc;
    __syncthreads();

    if (tid < 32) {
        const int w2 = tid >> 4, j2 = tid & 15;
        float s = 0.0f;
        for (int c = 0; c < 8; ++c) s += part[w2][j2][c];
        s += (w2 ? biasB[j2] : biasA[j2]);
        const float sig = 1.0f / (1.0f + __expf(-s));
        (w2 ? beta : alpha)[(size_t)m * NH_ + j2] = sig;
    }
}

// ---------------------------------------------------------------------------
// Per-head L2 normalization in place: one 64-wide head-row per wave32.
// Flat layout trick: row hr of [B*S, NH, D] starts at offset hr*64.
// ---------------------------------------------------------------------------
__global__ __launch_bounds__(256)
void l2norm_heads(float* __restrict__ t) {
    const int wave = threadIdx.x >> 5;
    const int lane = threadIdx.x & 31;
    const int hr   = blockIdx.x * 8 + wave;
    float* p = t + (size_t)hr * 64;
    float v0 = p[lane], v1 = p[lane + 32];
    float ss = v0 * v0 + v1 * v1;
    #pragma unroll
    for (int off = 16; off > 0; off >>= 1) ss += __shfl_xor(ss, off, 32);
    const float inv = 1.0f / fmaxf(sqrtf(ss), EPS_);
    p[lane]      = v0 * inv;
    p[lane + 32] = v1 * inv;
}

__global__ __launch_bounds__(256)
void tanh_inplace(float* __restrict__ p, int n) {
    const int i = blockIdx.x * 256 + threadIdx.x;
    if (i < n) p[i] = tanhf(p[i]);
}

// ---------------------------------------------------------------------------
// Sequential gated delta scan. One block per (b,h), 256 threads.
// State [64v x 64k] in registers: thread (vv = tid>>2, kg = tid&3) holds
// st[16] = state[vv][kg*16 .. kg*16+15]. The 4 kg-lanes of a vv are adjacent
// within a wave32 -> 2-step __shfl_xor butterfly does the k-reduction.
// Writes y to ws and final_state to d_out tail.
// ---------------------------------------------------------------------------
__global__ __launch_bounds__(256)
void scan_kernel(const float* __restrict__ q, const float* __restrict__ k,
                 const float* __restrict__ v, const float* __restrict__ alpha,
                 const float* __restrict__ beta,
                 const unsigned char* __restrict__ mask,
                 const float* __restrict__ state0,
                 float* __restrict__ y, float* __restrict__ fstate) {
    __shared__ float ks[64], qs[64], vs[64], ab[2];
    __shared__ int   msh;

    const int bh  = blockIdx.x;       // 0..31
    const int b   = bh >> 4;
    const int h   = bh & 15;
    const int tid = threadIdx.x;
    const int vv  = tid >> 2;         // 0..63
    const int kg  = tid & 3;          // k-group of 16

    float st[16];
    {
        const float* s0 = state0 + ((size_t)bh * 64 + vv) * 64 + kg * 16;
        #pragma unroll
        for (int j = 0; j < 16; ++j) st[j] = s0[j];
    }

    for (int t = 0; t < S_; ++t) {
        const size_t base = ((size_t)b * S_ + t) * HID_ + (size_t)h * 64;
        if (tid < 64)        ks[tid]       = k[base + tid];
        else if (tid < 128)  qs[tid - 64]  = q[base + tid - 64];
        else if (tid < 192)  vs[tid - 128] = v[base + tid - 128];
        else if (tid == 192) {
            const size_t g = (size_t)(b * S_ + t) * NH_ + h;
            ab[0] = alpha[g]; ab[1] = beta[g];
        } else if (tid == 193) {
            msh = (int)mask[b * S_ + t];
        }
        __syncthreads();

        const float a_t = ab[0];
        const float b_t = ab[1];
        const int   m   = msh;

        // read[vv] = sum_k state[vv][k] * k_t[k]
        float pr = 0.0f;
        #pragma unroll
        for (int j = 0; j < 16; ++j) pr += st[j] * ks[kg * 16 + j];
        pr += __shfl_xor(pr, 1, 32);
        pr += __shfl_xor(pr, 2, 32);          // broadcast across the 4 kg-lanes

        const float dv = vs[vv] - pr;         // (v - read)[vv]

        float py = 0.0f;
        #pragma unroll
        for (int j = 0; j < 16; ++j) {
            float cand = a_t * st[j] + b_t * dv * ks[kg * 16 + j];
            cand  = fminf(fmaxf(cand, -CLIP_), CLIP_);
            st[j] = m ? cand : st[j];
            py   += st[j] * qs[kg * 16 + j];
        }
        py += __shfl_xor(py, 1, 32);
        py += __shfl_xor(py, 2, 32);
        if (kg == 0) y[base + vv] = m ? py : 0.0f;
        __syncthreads();                      // protect ks/qs/vs for next step
    }

    float* fs = fstate + ((size_t)bh * 64 + vv) * 64 + kg * 16;
    #pragma unroll
    for (int j = 0; j < 16; ++j) fs[j] = st[j];
}

// ---------------------------------------------------------------------------
extern "C" void kernel_launch(void* const* d_in, const int* in_sizes, int n_in,
                              void* d_out, int out_size, void* d_ws, size_t ws_size,
                              hipStream_t stream) {
    const float* x      = (const float*)d_in[0];
    const float* Wq     = (const float*)d_in[1];
    const float* Wk     = (const float*)d_in[2];
    const float* Wv     = (const float*)d_in[3];
    const float* Wa     = (const float*)d_in[4];
    const float* ba     = (const float*)d_in[5];
    const float* Wb     = (const float*)d_in[6];
    const float* bbias  = (const float*)d_in[7];
    const float* Wo     = (const float*)d_in[8];
    const float* state0 = (const float*)d_in[9];
    const unsigned char* mask = (const unsigned char*)d_in[10];

    float* out    = (float*)d_out;                       // [B,S,HID]
    float* fstate = out + (size_t)B_ * S_ * HID_;        // [B,NH,D,D]

    const size_t NTOK = (size_t)B_ * S_;                 // 2048
    const size_t NELT = NTOK * HID_;                     // 2,097,152
    float* ws = (float*)d_ws;
    float* qb = ws;
    float* kb = qb + NELT;
    float* vb = kb + NELT;
    float* yb = vb + NELT;
    float* al = yb + NELT;
    float* be = al + NTOK * NH_;

    const int M = (int)NTOK, N = HID_, K = HID_;
    dim3 gg(N / 64, M / 128);

    gemm_f32_wmma<<<gg, 256, 0, stream>>>(x, Wq, qb, M, N, K);
    gemm_f32_wmma<<<gg, 256, 0, stream>>>(x, Wk, kb, M, N, K);
    gemm_f32_wmma<<<gg, 256, 0, stream>>>(x, Wv, vb, M, N, K);
    gates_kernel<<<M, 256, 0, stream>>>(x, Wa, ba, Wb, bbias, al, be);

    l2norm_heads<<<(M * NH_) / 8, 256, 0, stream>>>(qb);
    l2norm_heads<<<(M * NH_) / 8, 256, 0, stream>>>(kb);
    tanh_inplace<<<(int)(NELT / 256), 256, 0, stream>>>(vb, (int)NELT);

    scan_kernel<<<B_ * NH_, 256, 0, stream>>>(qb, kb, vb, al, be, mask, state0,
                                              yb, fstate);

    gemm_f32_wmma<<<gg, 256, 0, stream>>>(yb, Wo, out, M, N, K);
}